// Model_10514079940941
// MI455X (gfx1250) — compile-verified
//
#include <hip/hip_runtime.h>
#include <math.h>

// ---------------------------------------------------------------------------
// MI455X (gfx1250) implementation, round 2.
//
//   prep_kernel      : per-Gaussian conic (Sigma^-1), alpha*u table (1024 x 6)
//   pairwise_kernel  : 25,600 x 1,024 Gaussian field eval (VALU + v_exp_f32),
//                      24KB LDS broadcast table; emits padded conv input 1024x76
//   pack_conv/pack_l1: zero-padded, column-major weight packing so WMMA
//                      fragment loads are single aligned global_load_b64s
//   wmma_gemm_tanh   : fully-unrolled f32 GEMM on V_WMMA_F32_16X16X4_F32,
//                      8 wave32 tiles per 256-thread block, branch-free K loop,
//                      fused bias+tanh epilogue
//   emb_kernel       : tiny 7->30 embedding MLP (VALU)
//   heads_kernel     : 4 output heads + state update + new covariance
// ---------------------------------------------------------------------------

#define NG 1024
#define KS 5
#define NSAMP (NG * KS * KS)   // 25600

#define KC 76                  // conv K padded (75 -> 76)
#define NC 208                 // conv N padded (200 -> 208, 13 tiles)
#define KL 80                  // l1 K (already multiple of 4)
#define NL 80                  // l1 N (5 tiles)

typedef float v2f __attribute__((ext_vector_type(2)));
typedef float v8f __attribute__((ext_vector_type(8)));

// --------------------------- prep: conics table ----------------------------
__global__ __launch_bounds__(256) void prep_kernel(
    const float* __restrict__ means, const float* __restrict__ u,
    const float* __restrict__ scaling, const float* __restrict__ transform,
    const float* __restrict__ opac, float* __restrict__ table)
{
    int n = blockIdx.x * blockDim.x + threadIdx.x;
    if (n >= NG) return;
    float s0 = scaling[2 * n], s1 = scaling[2 * n + 1], t = transform[n];
    float c00 = s0 * s0, c01 = s0 * t, c11 = t * t + s1 * s1;
    float det = c00 * c11 - c01 * c01;
    float inv = 1.0f / det;
    float iA = c11 * inv, iB = -c01 * inv, iC = c00 * inv;
    float alpha = 1.0f / (1.0f + __expf(-opac[n]));
    float* tb = table + n * 6;
    tb[0] = iA; tb[1] = iB; tb[2] = iC;
    tb[3] = means[2 * n]; tb[4] = means[2 * n + 1];
    tb[5] = alpha * u[n];
}

// ------------------ pairwise Gaussian field + Laplacian --------------------
__global__ __launch_bounds__(256) void pairwise_kernel(
    const float* __restrict__ table, const float* __restrict__ means,
    float* __restrict__ Aconv /* NG x KC, channel-major, col 75 zero pad */)
{
    __shared__ float sh[NG * 6];    // 24 KB of the 320 KB/WGP LDS
    __builtin_prefetch(table + threadIdx.x * 16, 0, 3);
    for (int i = threadIdx.x; i < NG * 6; i += blockDim.x) sh[i] = table[i];
    __syncthreads();

    int m = blockIdx.x * blockDim.x + threadIdx.x;   // 0..25599 exactly
    int g = m / 25, s = m - g * 25;
    int i = s / 5, j = s - i * 5;
    float sx = means[2 * g]     + (float)(i - 2) * 0.01f;
    float sy = means[2 * g + 1] + (float)(j - 2) * 0.01f;
    float bc = (fabsf(sx) < 1.0f && fabsf(sy) < 1.0f) ? 1.0f : 0.0f;

    float us = 0.0f, lp = 0.0f;
#pragma unroll 4
    for (int n = 0; n < NG; ++n) {
        const float* tb = &sh[n * 6];           // all lanes same addr: broadcast
        float A = tb[0], B = tb[1], C = tb[2];
        float dx = sx - tb[3], dy = sy - tb[4];
        float au = tb[5];
        float cx = fmaf(A, dx, B * dy);         // (Sigma^-1 d)_x
        float cy = fmaf(B, dx, C * dy);         // (Sigma^-1 d)_y
        float q  = fmaf(dx, cx, dy * cy);       // d^T Sigma^-1 d
        float e  = __expf(-0.5f * q) * au;      // G * u  (alpha folded into au)
        float S  = fmaf(cx, cx, cy * cy) - (A + C);
        us += e;
        lp = fmaf(e, S, lp);
    }
    // img layout: [u_s(25) | lap(25) | bc(25) | 0] per gaussian
    Aconv[g * KC +  0 + s] = us;
    Aconv[g * KC + 25 + s] = lp;
    Aconv[g * KC + 50 + s] = bc;
    if (s == 0) Aconv[g * KC + 75] = 0.0f;      // K pad
}

// --------------------- weight packing (zero-padded, col-major) -------------
// Bc[col][k] = convW_flat[col*75 + k]  (col < 200, k < 75), else 0.
__global__ __launch_bounds__(256) void pack_conv(
    const float* __restrict__ convW, const float* __restrict__ convb,
    float* __restrict__ Bc, float* __restrict__ biasc)
{
    int idx = blockIdx.x * blockDim.x + threadIdx.x;
    if (idx < NC * KC) {
        int col = idx / KC, k = idx - col * KC;
        Bc[idx] = (col < 200 && k < 75) ? convW[col * 75 + k] : 0.0f;
    }
    if (idx < NC) biasc[idx] = (idx < 200) ? convb[idx] : 0.0f;
}

// Bl[b][col][k] = l1W[b][k][col]  (transpose to col-major per batch)
__global__ __launch_bounds__(256) void pack_l1(
    const float* __restrict__ l1W, float* __restrict__ Bl)
{
    int idx = blockIdx.x * blockDim.x + threadIdx.x;
    if (idx >= 4 * NL * KL) return;
    int b = idx / (NL * KL), r = idx - b * (NL * KL);
    int col = r / KL, k = r - col * KL;
    Bl[idx] = l1W[b * (NL * KL) + k * NL + col];
}

// ----------------- f32 WMMA GEMM with fused bias + tanh --------------------
// A: M x KD row-major (K-padded).  Bc: Ncols x KD column-packed (so the
// B-fragment pair (k0,k0+1) for a column is one aligned b64 load).
// One wave32 per 16x16 tile; 8 tiles per 256-thread block; batch = blockIdx.y.
// KD is a template constant -> fully unrolled, branch-free K loop.
template<int KD, int MODE>
__global__ __launch_bounds__(256) void wmma_gemm_tanh(
    const float* __restrict__ A,  long sAb,
    const float* __restrict__ Bc, long sBb,
    const float* __restrict__ bias, int sBiasb,
    float* __restrict__ out, long sOutb,
    int M, int nTilesN, int NcolsValid)
{
    int wave = threadIdx.x >> 5;
    int tile = blockIdx.x * 8 + wave;
    int tilesM = M >> 4;
    if (tile >= tilesM * nTilesN) return;      // wave-uniform: EXEC stays all-1
    int tm = tile % tilesM;
    int tn = tile / tilesM;
    int batch = blockIdx.y;
    A    += (long)batch * sAb;
    Bc   += (long)batch * sBb;
    bias += batch * sBiasb;
    out  += (long)batch * sOutb;

    int lane = threadIdx.x & 31;
    int half = lane >> 4;       // 0: K=0,1 / rows 0-7 ; 1: K=2,3 / rows 8-15
    int lr   = lane & 15;
    const float* Ap = A  + (long)(tm * 16 + lr) * KD + 2 * half;
    const float* Bp = Bc + (long)(tn * 16 + lr) * KD + 2 * half;

    v8f acc = {0.f, 0.f, 0.f, 0.f, 0.f, 0.f, 0.f, 0.f};
#pragma unroll
    for (int kb = 0; kb < KD; kb += 4) {
        v2f a = *(const v2f*)(Ap + kb);        // aligned global_load_b64
        v2f b = *(const v2f*)(Bp + kb);        // aligned global_load_b64
        acc = __builtin_amdgcn_wmma_f32_16x16x4_f32(
            false, a, false, b, (short)0, acc, false, false);
    }

    int ocol = tn * 16 + lr;
    float bz = bias[ocol];
#pragma unroll
    for (int r = 0; r < 8; ++r) {
        int orow = tm * 16 + r + 8 * half;     // C/D: lanes 16-31 hold M=r+8
        float v = tanhf(acc[r] + bz);
        if (MODE == 0) {                       // conv -> z[k][n][o], o < 50
            if (ocol < NcolsValid) {
                int kk = ocol / 50, oo = ocol - kk * 50;
                out[((long)kk * M + orow) * 80 + oo] = v;
            }
        } else {                               // l1 -> h[n][80]
            out[(long)orow * 80 + ocol] = v;
        }
    }
}

// --------------------------- embedding branch ------------------------------
__global__ __launch_bounds__(256) void emb_kernel(
    const float* __restrict__ means, const float* __restrict__ u,
    const float* __restrict__ scaling, const float* __restrict__ transform,
    const float* __restrict__ opac,
    const float* __restrict__ embW, const float* __restrict__ embb,
    float* __restrict__ z)
{
    int id = blockIdx.x * blockDim.x + threadIdx.x;   // k*NG + n
    if (id >= 4 * NG) return;
    int k = id / NG, n = id - k * NG;
    float p[7] = { means[2 * n], means[2 * n + 1], u[n],
                   scaling[2 * n], scaling[2 * n + 1], transform[n], opac[n] };
    const float* W = embW + k * 7 * 30;
    const float* b = embb + k * 30;
    float* zo = z + ((long)k * NG + n) * 80 + 50;
#pragma unroll 5
    for (int g = 0; g < 30; ++g) {
        float acc = b[g];
#pragma unroll
        for (int f = 0; f < 7; ++f) acc = fmaf(p[f], W[f * 30 + g], acc);
        zo[g] = tanhf(acc);
    }
}

// ------------------------- output heads + update ---------------------------
__global__ __launch_bounds__(256) void heads_kernel(
    const float* __restrict__ h, const float* __restrict__ means,
    const float* __restrict__ u, const float* __restrict__ scaling,
    const float* __restrict__ transform,
    const float* __restrict__ WS,  const float* __restrict__ bS,
    const float* __restrict__ WT,  const float* __restrict__ bT,
    const float* __restrict__ WSc, const float* __restrict__ bSc,
    const float* __restrict__ WTf, const float* __restrict__ bTf,
    float* __restrict__ out)
{
    int n = blockIdx.x * blockDim.x + threadIdx.x;
    if (n >= NG) return;
    const float* h0 = h + ((long)0 * NG + n) * 80;
    const float* h1 = h + ((long)1 * NG + n) * 80;
    const float* h2 = h + ((long)2 * NG + n) * 80;
    const float* h3 = h + ((long)3 * NG + n) * 80;
    float d  = bS[0];
    float t0 = bT[0],  t1 = bT[1];
    float c0 = bSc[0], c1 = bSc[1];
    float tf = bTf[0];
#pragma unroll 4
    for (int f = 0; f < 80; ++f) {
        d  = fmaf(h0[f], WS[f],          d);
        t0 = fmaf(h1[f], WT[f * 2 + 0],  t0);
        t1 = fmaf(h1[f], WT[f * 2 + 1],  t1);
        c0 = fmaf(h2[f], WSc[f * 2 + 0], c0);
        c1 = fmaf(h2[f], WSc[f * 2 + 1], c1);
        tf = fmaf(h3[f], WTf[f],         tf);
    }
    float un = u[n] + d;
    float mx = means[2 * n] + t0, my = means[2 * n + 1] + t1;
    float s0 = scaling[2 * n] * __expf(c0);
    float s1 = scaling[2 * n + 1] * __expf(c1);
    float tn = transform[n] + tf;
    float c00 = s0 * s0, c01 = s0 * tn, c11 = tn * tn + s1 * s1;
    out[n] = un;                                   // u_new       (N,1)
    out[NG + 2 * n]     = mx;                      // means_new   (N,2)
    out[NG + 2 * n + 1] = my;
    out[3 * NG + 2 * n]     = s0;                  // scaling_new (N,2)
    out[3 * NG + 2 * n + 1] = s1;
    out[5 * NG + n] = tn;                          // transform_new (N,1)
    out[6 * NG + 4 * n + 0] = c00;                 // cov_new     (N,2,2)
    out[6 * NG + 4 * n + 1] = c01;
    out[6 * NG + 4 * n + 2] = c01;
    out[6 * NG + 4 * n + 3] = c11;
}

// ------------------------------- launcher ----------------------------------
extern "C" void kernel_launch(void* const* d_in, const int* in_sizes, int n_in,
                              void* d_out, int out_size, void* d_ws, size_t ws_size,
                              hipStream_t stream)
{
    const float* means     = (const float*)d_in[0];
    const float* u         = (const float*)d_in[1];
    const float* scaling   = (const float*)d_in[2];
    const float* transform = (const float*)d_in[3];
    const float* opac      = (const float*)d_in[4];
    const float* convW     = (const float*)d_in[5];   // (4,50,3,5,5)
    const float* convb     = (const float*)d_in[6];   // (4,50)
    const float* embW      = (const float*)d_in[7];   // (4,7,30)
    const float* embb      = (const float*)d_in[8];   // (4,30)
    const float* l1W       = (const float*)d_in[9];   // (4,80,80)
    const float* l1b       = (const float*)d_in[10];  // (4,80)
    const float* WS        = (const float*)d_in[11];
    const float* bS        = (const float*)d_in[12];
    const float* WT        = (const float*)d_in[13];
    const float* bT        = (const float*)d_in[14];
    const float* WSc       = (const float*)d_in[15];
    const float* bSc       = (const float*)d_in[16];
    const float* WTf       = (const float*)d_in[17];
    const float* bTf       = (const float*)d_in[18];

    float* out = (float*)d_out;
    float* ws  = (float*)d_ws;
    float* table = ws;                       // 1024*6        =   6,144 floats
    float* Aconv = table + NG * 6;           // 1024*76       =  77,824
    float* BcC   = Aconv + NG * KC;          // 208*76        =  15,808
    float* biasC = BcC + NC * KC;            // 208
    float* Bl    = biasC + NC;               // 4*80*80       =  25,600
    float* z     = Bl + 4 * NL * KL;         // 4*1024*80     = 327,680
    float* h     = z + 4 * NG * 80;          // 4*1024*80     = 327,680 (~3.1MB)

    prep_kernel<<<(NG + 255) / 256, 256, 0, stream>>>(
        means, u, scaling, transform, opac, table);

    pack_conv<<<(NC * KC + 255) / 256, 256, 0, stream>>>(convW, convb, BcC, biasC);
    pack_l1<<<(4 * NL * KL + 255) / 256, 256, 0, stream>>>(l1W, Bl);

    pairwise_kernel<<<NSAMP / 256, 256, 0, stream>>>(table, means, Aconv);

    // conv: (1024 x 76) @ (76 x 208) -> tanh -> z[k][n][0..49];  832 tiles
    wmma_gemm_tanh<KC, 0><<<dim3((64 * 13 + 7) / 8, 1), 256, 0, stream>>>(
        Aconv, 0, BcC, 0, biasC, 0, z, 0, NG, 13, 200);

    emb_kernel<<<(4 * NG + 255) / 256, 256, 0, stream>>>(
        means, u, scaling, transform, opac, embW, embb, z);

    // l1 (batched over k): (1024 x 80) @ (80 x 80) -> tanh -> h;  320 tiles/batch
    wmma_gemm_tanh<KL, 1><<<dim3((64 * 5 + 7) / 8, 4), 256, 0, stream>>>(
        z, (long)NG * 80, Bl, (long)NL * KL, l1b, 80, h, (long)NG * 80, NG, 5, 80);

    heads_kernel<<<(NG + 255) / 256, 256, 0, stream>>>(
        h, means, u, scaling, transform, WS, bS, WT, bT, WSc, bSc, WTf, bTf, out);
}